// PieceVectorExtractor_19061064860376
// MI455X (gfx1250) — compile-verified
//
#include <hip/hip_runtime.h>
#include <hip/hip_bf16.h>
#include <stdint.h>

// Problem constants (from reference): B=1024, C=512, H*W=S=64, 32 pieces.
#define NUM_PIECES 32
#define C_TOTAL    512
#define S_TOTAL    64
#define CC         128                 // channels staged per workgroup
#define CHUNKS     (C_TOTAL / CC)      // 4
#define LDS_ROW    68                  // 64 floats + 16B pad: keeps B128 alignment,
                                       // transposed-gather bank conflicts capped at 2-way
#define THREADS    256                 // 8 wave32 waves

#if defined(__HIP_DEVICE_COMPILE__) && __has_builtin(__builtin_amdgcn_global_load_async_to_lds_b128)
#define HAVE_ASYNC_LDS 1
#else
#define HAVE_ASYNC_LDS 0
#endif

typedef int v4i __attribute__((vector_size(16)));
typedef __attribute__((address_space(1))) v4i* glob_v4i_ptr;   // HIP "__device__" AS
typedef __attribute__((address_space(3))) v4i* lds_v4i_ptr;    // HIP "__shared__" AS

__global__ __launch_bounds__(THREADS)
void piece_gather_kernel(const float* __restrict__ fbv,
                         const int*   __restrict__ ids,
                         float*       __restrict__ out)
{
    __shared__ float tile[CC * LDS_ROW];   // ~34 KB
    __shared__ int   sIdx[NUM_PIECES];

    const int b   = blockIdx.y;
    const int c0  = blockIdx.x * CC;
    const int tid = threadIdx.x;

    // ---- Step 1: kick off the async global->LDS DMA for fbv[b, c0:c0+CC, :]
    // 128 rows x 256 B/row = 2048 B128 transfers; 8 per thread. Global side is
    // fully contiguous (512 B per wave per issue); LDS side is row-padded.
    const float* gbase = fbv + ((size_t)b * C_TOTAL + c0) * S_TOTAL;

#if HAVE_ASYNC_LDS
#pragma unroll
    for (int k = 0; k < (CC * S_TOTAL) / (4 * THREADS); ++k) {
        const int t   = tid + k * THREADS;
        const int row = t >> 4;        // channel within chunk
        const int col = t & 15;        // 16B segment within row
        __builtin_amdgcn_global_load_async_to_lds_b128(
            (glob_v4i_ptr)(gbase + row * S_TOTAL + col * 4),
            (lds_v4i_ptr)(&tile[row * LDS_ROW + col * 4]),
            0, 0);
    }
#else
    // Fallback: plain vector load + LDS store (still coalesced, just synchronous)
#pragma unroll
    for (int k = 0; k < (CC * S_TOTAL) / (4 * THREADS); ++k) {
        const int t   = tid + k * THREADS;
        const int row = t >> 4;
        const int col = t & 15;
        const float4 v = *(const float4*)(gbase + row * S_TOTAL + col * 4);
        *(float4*)(&tile[row * LDS_ROW + col * 4]) = v;
    }
#endif

    // ---- Step 2: branchless first-index scan, overlapped with the DMA ----
    // Reverse iteration: last overwrite wins -> smallest matching s (argmax of
    // the row-major mask, matching the reference). Uniform loads -> s_load
    // batches; the select chain is pure v_cndmask, no branches.
    if (tid < NUM_PIECES) {
        const int target = tid + 1;
        const int* idb = ids + b * S_TOTAL;
        int first = -1;
#pragma unroll
        for (int s = S_TOTAL - 1; s >= 0; --s) {
            first = (idb[s] == target) ? s : first;
        }
        sIdx[tid] = first;
    }

    // ---- Step 3: drain the DMA, then rendezvous ----
#if HAVE_ASYNC_LDS
#if __has_builtin(__builtin_amdgcn_s_wait_asynccnt)
    __builtin_amdgcn_s_wait_asynccnt(0);
#else
    asm volatile("s_wait_asynccnt 0" ::: "memory");
#endif
#endif
    __syncthreads();

    // ---- Step 4: coalesced writes out[b, p, c0+cc] from LDS gather ----
    // e = p*CC + cc; consecutive lanes -> consecutive cc -> 128B contiguous
    // stores per wave; sIdx[p] is wave-uniform within each iteration.
    float* ob = out + ((size_t)b * NUM_PIECES) * C_TOTAL + c0;
#pragma unroll
    for (int k = 0; k < (NUM_PIECES * CC) / THREADS; ++k) {
        const int e  = tid + k * THREADS;
        const int p  = e >> 7;         // e / CC
        const int cc = e & (CC - 1);   // e % CC
        const int s  = sIdx[p];
        const float v = (s >= 0) ? tile[cc * LDS_ROW + s] : 0.0f;
        ob[p * C_TOTAL + cc] = v;
    }
}

extern "C" void kernel_launch(void* const* d_in, const int* in_sizes, int n_in,
                              void* d_out, int out_size, void* d_ws, size_t ws_size,
                              hipStream_t stream) {
    const float* fbv = (const float*)d_in[0];   // (B, 512, 8, 8) f32
    const int*   ids = (const int*)d_in[1];     // (B, 8, 8) int (per harness contract)
    float*       out = (float*)d_out;           // (B, 32, 512) f32

    const int B = in_sizes[1] / S_TOTAL;        // 1024
    dim3 grid(CHUNKS, B);
    hipLaunchKernelGGL(piece_gather_kernel, grid, dim3(THREADS), 0, stream,
                       fbv, ids, out);
}